// Upsample_22514218566128
// MI455X (gfx1250) — compile-verified
//
#include <hip/hip_runtime.h>

// upfirdn2d(x, k, up=2, down=1, pad=(2,1)):  (B,C,64,64) f32 -> (B,C,128,128) f32
// Memory-bound (AI ~ 1.6 flop/byte; 64 MiB in + 256 MiB out @ 23.3 TB/s ~ 14 us).
// CDNA5 data path:
//   - async global->LDS staging of each 16KB input plane (ASYNCcnt), single base
//     address + INST_OFFSET for all 4 transfers (offset applies to LDS and global)
//   - th:TH_LOAD_NT on staging loads, TH_STORE_NT on output stores: the 256MB
//     streaming output exceeds the 192MB L2, so bypassing temporal caching avoids
//     thrash; nothing in this kernel is ever re-read from L2.
//   - LDS-resident 2x2-tap stencil with 4x spatial reuse
//   - one float4 per lane => 512B coalesced stores per wave

#define IN_H   64
#define IN_W   64
#define OUT_H  128
#define OUT_W  128
#define PLANE_IN  (IN_H * IN_W)     // 4096 floats = 16 KB
#define PLANE_OUT (OUT_H * OUT_W)   // 16384 floats = 64 KB

typedef float v4f __attribute__((ext_vector_type(4)));   // native vector: OK for
                                                         // __builtin_nontemporal_store

__global__ __launch_bounds__(256) void upfirdn2d_up2_kernel(
    const float* __restrict__ x,      // (planes, 64, 64)
    const float* __restrict__ k,      // (4, 4) FIR kernel
    float* __restrict__ out)          // (planes, 128, 128)
{
    __shared__ float s_in[PLANE_IN];  // whole input plane, 16 KB

    const int tid   = threadIdx.x;
    const int plane = blockIdx.x;
    const float* src  = x   + (size_t)plane * PLANE_IN;
    float*       dstp = out + (size_t)plane * PLANE_OUT;

    // ---- Async DMA of the 64x64 plane into LDS: 256 thr * 4 * 16B = 16 KB ----
    // LDS address operand = low 32 bits of the generic shared pointer
    // (flat addressing). INST_OFFSET is added to both
    // the LDS and the global address, so one base VGPR set serves all 4 issues.
    {
        const int elt = tid * 4;                          // float index, 16B aligned
        unsigned lds_addr =
            (unsigned)(unsigned long long)(const void*)(&s_in[elt]);
        unsigned long long gaddr =
            (unsigned long long)(const void*)(src + elt);
#pragma unroll
        for (int p = 0; p < 4; ++p) {
            asm volatile(
                "global_load_async_to_lds_b128 %0, %1, off offset:%2 th:TH_LOAD_NT"
                :: "v"(lds_addr), "v"(gaddr), "i"(p * 4096)
                : "memory");
        }
    }
    asm volatile("s_wait_asynccnt 0" ::: "memory");       // my async writes landed in LDS
    __syncthreads();                                      // publish to the other 7 waves

    // ---- Output mapping: lane txq of a wave -> float4 at ox0 = 4*txq on row oy ----
    const int txq  = tid & 31;    // 32 float4 per 128-wide output row (one wave/row)
    const int trow = tid >> 5;    // 8 rows per pass, 16 passes
    const int ry   = trow & 1;    // row phase is invariant (row stride 8 is even)
    const int m0   = txq * 2;     // input column base; needs cols m0-1 .. m0+2

    // Kernel weights for this row phase (uniform per thread, tiny L2-resident load).
    // Even output rows use kernel rows (3,1); odd rows use (2,0). Same for columns.
    const float wt0 = k[(3 - ry) * 4 + 0];
    const float wt1 = k[(3 - ry) * 4 + 1];
    const float wt2 = k[(3 - ry) * 4 + 2];
    const float wt3 = k[(3 - ry) * 4 + 3];
    const float wb0 = k[(1 - ry) * 4 + 0];
    const float wb1 = k[(1 - ry) * 4 + 1];
    const float wb2 = k[(1 - ry) * 4 + 2];
    const float wb3 = k[(1 - ry) * 4 + 3];

    const bool c0ok = (m0 - 1) >= 0;      // left halo column exists
    const bool c3ok = (m0 + 2) < IN_W;    // right halo column exists

#pragma unroll 4
    for (int it = 0; it < 16; ++it) {
        const int oy = trow + it * 8;
        const int my = oy >> 1;
        const int yT = my + ry - 1;       // top input row of the 2-row window
        const int yB = yT + 1;
        const bool topok = yT >= 0;
        const bool botok = yB < IN_H;

        const int rT = yT * IN_W;
        const int rB = yB * IN_W;

        const float t0 = (topok && c0ok) ? s_in[rT + m0 - 1] : 0.f;
        const float t1 = topok           ? s_in[rT + m0    ] : 0.f;
        const float t2 = topok           ? s_in[rT + m0 + 1] : 0.f;
        const float t3 = (topok && c3ok) ? s_in[rT + m0 + 2] : 0.f;
        const float b0 = (botok && c0ok) ? s_in[rB + m0 - 1] : 0.f;
        const float b1 = botok           ? s_in[rB + m0    ] : 0.f;
        const float b2 = botok           ? s_in[rB + m0 + 1] : 0.f;
        const float b3 = (botok && c3ok) ? s_in[rB + m0 + 2] : 0.f;

        // 4 consecutive output pixels; column phases alternate 0,1,0,1.
        v4f o;
        o.x = wt3 * t0 + wt1 * t1 + wb3 * b0 + wb1 * b1;   // rx=0, xL=m0-1
        o.y = wt2 * t1 + wt0 * t2 + wb2 * b1 + wb0 * b2;   // rx=1, xL=m0
        o.z = wt3 * t1 + wt1 * t2 + wb3 * b1 + wb1 * b2;   // rx=0, xL=m0
        o.w = wt2 * t2 + wt0 * t3 + wb2 * b2 + wb0 * b3;   // rx=1, xL=m0+1

        // Streaming store: output (256MB) > L2 (192MB), never re-read -> NT.
        __builtin_nontemporal_store(o, (v4f*)(dstp + oy * OUT_W + txq * 4));
    }
}

extern "C" void kernel_launch(void* const* d_in, const int* in_sizes, int n_in,
                              void* d_out, int out_size, void* d_ws, size_t ws_size,
                              hipStream_t stream) {
    (void)n_in; (void)out_size; (void)d_ws; (void)ws_size;
    const float* x = (const float*)d_in[0];      // (16,256,64,64) f32
    const float* k = (const float*)d_in[1];      // (4,4) f32
    float* out = (float*)d_out;                  // (16,256,128,128) f32

    const int planes = in_sizes[0] / PLANE_IN;   // 16*256 = 4096
    upfirdn2d_up2_kernel<<<dim3(planes), dim3(256), 0, stream>>>(x, k, out);
}